// RelationalGAT_75831942578727
// MI455X (gfx1250) — compile-verified
//
#include <hip/hip_runtime.h>
#include <hip/hip_bf16.h>
#include <math.h>

// ---------------------------------------------------------------------------
// GAT (2-layer, PyG GATConv semantics) for gfx1250 (MI455X).
//   * Dense projections via V_WMMA_F32_16X16X4_F32 (exact fp32 numerics),
//     shapes templated so the K-loop is fully unrolled with immediate-offset
//     loads (no per-iteration 64-bit address math).
//   * Edge softmax via order-preserving-uint atomicMax + atomicAdd passes.
//   * Wave-per-edge (wave32) weighted scatter-add, 2 channels per lane.
// ---------------------------------------------------------------------------

typedef __attribute__((ext_vector_type(2))) float v2f;
typedef __attribute__((ext_vector_type(8))) float v8f;

#define NEG_SLOPE 0.2f

// ---- order-preserving float <-> uint map (for atomicMax-based segment max) --
__device__ __forceinline__ unsigned fkey(float f) {
  unsigned u = __float_as_uint(f);
  return (u & 0x80000000u) ? ~u : (u | 0x80000000u);
}
__device__ __forceinline__ float funkey(unsigned k) {
  return __uint_as_float((k & 0x80000000u) ? (k & 0x7FFFFFFFu) : ~k);
}

// ---------------------------------------------------------------------------
// C[M x NCOLS] = A[M x K] @ B[K x NCOLS], fp32, via WMMA 16x16x4.
// One wave per (16-row strip, 64-col tile): 4 accumulators, A frag reused.
// K, NCOLS compile-time => fully unrolled, all loads base+imm-offset.
// Block = 128 threads = 4 waves (4 consecutive strips).
// ---------------------------------------------------------------------------
template <int K, int NCOLS>
__global__ void gat_gemm_f32_wmma(const float* __restrict__ A,
                                  const float* __restrict__ B,
                                  float* __restrict__ C, int M) {
  const int lane  = threadIdx.x & 31;
  const int wave  = threadIdx.x >> 5;
  const int strip = blockIdx.y * 4 + wave;     // 16-row strip index
  if (strip * 16 >= M) return;

  const int row   = strip * 16 + (lane & 15);  // A-fragment row for this lane
  const int kHalf = (lane >> 4) << 1;          // lanes 0-15 -> K+0/1, 16-31 -> K+2/3
  const int ncol  = blockIdx.x * 64 + (lane & 15);

  v8f acc0 = {}, acc1 = {}, acc2 = {}, acc3 = {};
  const float* __restrict__ Ap = A + (size_t)row * K + kHalf;
  const float* __restrict__ Bp = B + (size_t)kHalf * NCOLS + ncol;

#pragma unroll
  for (int k0 = 0; k0 < K; k0 += 4) {
    // One speculative prefetch per A cacheline (emits global_prefetch_b8).
    if ((k0 & 63) == 0) __builtin_prefetch(Ap + k0 + 128, 0, 1);

    // A fragment: VGPR0 = A[row][k0+kHalf], VGPR1 = A[row][k0+kHalf+1]
    v2f a = *(const v2f*)(Ap + k0);

    // B fragment: VGPR0 = B[k0+kHalf][n], VGPR1 = B[k0+kHalf+1][n]
    const float* __restrict__ bp = Bp + (size_t)k0 * NCOLS;
    v2f b0 = { bp[ 0], bp[NCOLS +  0] };
    v2f b1 = { bp[16], bp[NCOLS + 16] };
    v2f b2 = { bp[32], bp[NCOLS + 32] };
    v2f b3 = { bp[48], bp[NCOLS + 48] };

    acc0 = __builtin_amdgcn_wmma_f32_16x16x4_f32(false, a, false, b0, (short)0, acc0, false, false);
    acc1 = __builtin_amdgcn_wmma_f32_16x16x4_f32(false, a, false, b1, (short)0, acc1, false, false);
    acc2 = __builtin_amdgcn_wmma_f32_16x16x4_f32(false, a, false, b2, (short)0, acc2, false, false);
    acc3 = __builtin_amdgcn_wmma_f32_16x16x4_f32(false, a, false, b3, (short)0, acc3, false, false);
  }

  // C/D layout: VGPR r -> rows r (lanes 0-15) / r+8 (lanes 16-31), N = lane%16.
  const int rowHi = (lane >> 4) * 8;
#pragma unroll
  for (int r = 0; r < 8; ++r) {
    float* __restrict__ Crow = C + (size_t)(strip * 16 + rowHi + r) * NCOLS;
    Crow[ncol +  0] = acc0[r];
    Crow[ncol + 16] = acc1[r];
    Crow[ncol + 32] = acc2[r];
    Crow[ncol + 48] = acc3[r];
  }
}

// ---------------------------------------------------------------------------
// Per-node attention logits: al_s[n,h] = <h[n,h,:], att_s[h,:]>, same for d.
// ---------------------------------------------------------------------------
__global__ void gat_attn_dot(const float* __restrict__ h,
                             const float* __restrict__ att_s,
                             const float* __restrict__ att_d,
                             float* __restrict__ al_s,
                             float* __restrict__ al_d,
                             int N, int H, int C) {
  int idx = blockIdx.x * blockDim.x + threadIdx.x;
  if (idx >= N * H) return;
  int n = idx / H, hh = idx - n * H;
  const float* __restrict__ hp = h + (size_t)n * H * C + hh * C;
  const float* __restrict__ sp = att_s + hh * C;
  const float* __restrict__ dp = att_d + hh * C;
  float ss = 0.f, sd = 0.f;
  for (int c = 0; c < C; ++c) {
    float v = hp[c];
    ss = fmaf(v, sp[c], ss);
    sd = fmaf(v, dp[c], sd);
  }
  al_s[idx] = ss;
  al_d[idx] = sd;
}

__device__ __forceinline__ void edge_nodes(const int* __restrict__ ei, int E,
                                           int e, int& s, int& d) {
  if (e < E) { s = ei[e]; d = ei[E + e]; }
  else       { s = d = e - E; }            // implicit self-loops
}

// ---------------------------------------------------------------------------
// Pass 1: segment max of leaky_relu(al_s[src]+al_d[dst]) into mkey[dst,h].
// ---------------------------------------------------------------------------
__global__ void gat_edge_max(const int* __restrict__ ei, int E, int Etot,
                             const float* __restrict__ al_s,
                             const float* __restrict__ al_d,
                             unsigned* __restrict__ mkey, int H) {
  int t = blockIdx.x * blockDim.x + threadIdx.x;
  if (t >= Etot * H) return;
  int e = t / H, hh = t - e * H;
  int s, d; edge_nodes(ei, E, e, s, d);
  float v = al_s[s * H + hh] + al_d[d * H + hh];
  v = v > 0.f ? v : NEG_SLOPE * v;
  atomicMax(&mkey[d * H + hh], fkey(v));
}

// ---------------------------------------------------------------------------
// Pass 2: denom[dst,h] += exp(e - m[dst,h]).
// ---------------------------------------------------------------------------
__global__ void gat_edge_expsum(const int* __restrict__ ei, int E, int Etot,
                                const float* __restrict__ al_s,
                                const float* __restrict__ al_d,
                                const unsigned* __restrict__ mkey,
                                float* __restrict__ denom, int H) {
  int t = blockIdx.x * blockDim.x + threadIdx.x;
  if (t >= Etot * H) return;
  int e = t / H, hh = t - e * H;
  int s, d; edge_nodes(ei, E, e, s, d);
  float v = al_s[s * H + hh] + al_d[d * H + hh];
  v = v > 0.f ? v : NEG_SLOPE * v;
  float m = funkey(mkey[d * H + hh]);
  atomicAdd(&denom[d * H + hh], __expf(v - m));
}

// ---------------------------------------------------------------------------
// Pass 3: agg[dst] += h[src] * alpha. One wave32 per edge; 64 channels/head
// handled as 2 per lane. alpha recomputed (cheap, broadcast loads).
// C is fixed at 64 for both layers.
// ---------------------------------------------------------------------------
__global__ void gat_edge_aggregate(const int* __restrict__ ei, int E, int Etot,
                                   const float* __restrict__ al_s,
                                   const float* __restrict__ al_d,
                                   const unsigned* __restrict__ mkey,
                                   const float* __restrict__ denom,
                                   const float* __restrict__ hsrc,
                                   float* __restrict__ agg, int H) {
  int lane = threadIdx.x & 31;
  int w = (blockIdx.x * blockDim.x + threadIdx.x) >> 5;
  if (w >= Etot) return;
  int s, d; edge_nodes(ei, E, w, s, d);
  const int HC = H * 64;
  for (int hh = 0; hh < H; ++hh) {
    float v = al_s[s * H + hh] + al_d[d * H + hh];
    v = v > 0.f ? v : NEG_SLOPE * v;
    float m = funkey(mkey[d * H + hh]);
    float alpha = __expf(v - m) / (denom[d * H + hh] + 1e-16f);
    const float* __restrict__ hp = hsrc + (size_t)s * HC + hh * 64;
    float* __restrict__ op = agg + (size_t)d * HC + hh * 64;
    atomicAdd(&op[lane],      hp[lane]      * alpha);
    atomicAdd(&op[lane + 32], hp[lane + 32] * alpha);
  }
}

// ---------------------------------------------------------------------------
// buf[t] = act(buf[t] + bias[t % Cb]);  act = ELU if doElu, identity otherwise
// ---------------------------------------------------------------------------
__global__ void gat_bias_act(float* __restrict__ buf, const float* __restrict__ bias,
                             int total, int Cb, int doElu) {
  int t = blockIdx.x * blockDim.x + threadIdx.x;
  if (t >= total) return;
  float v = buf[t] + bias[t % Cb];
  if (doElu) v = v > 0.f ? v : expm1f(v);
  buf[t] = v;
}

__global__ void gat_zero_f32(float* __restrict__ p, int n) {
  int t = blockIdx.x * blockDim.x + threadIdx.x;
  if (t < n) p[t] = 0.f;
}

// ---------------------------------------------------------------------------

static inline unsigned cdiv(unsigned a, unsigned b) { return (a + b - 1) / b; }

extern "C" void kernel_launch(void* const* d_in, const int* in_sizes, int n_in,
                              void* d_out, int out_size, void* d_ws, size_t ws_size,
                              hipStream_t stream) {
  const float* x   = (const float*)d_in[0];
  const int*   ei  = (const int*)  d_in[1];
  const float* W1  = (const float*)d_in[2];
  const float* as1 = (const float*)d_in[3];
  const float* ad1 = (const float*)d_in[4];
  const float* b1  = (const float*)d_in[5];
  const float* W2  = (const float*)d_in[6];
  const float* as2 = (const float*)d_in[7];
  const float* ad2 = (const float*)d_in[8];
  const float* b2  = (const float*)d_in[9];
  float* out = (float*)d_out;

  const int IN  = 256;          // in_dim
  const int H   = 4;            // heads, layer 1
  const int C   = 64;           // per-head channels (both layers)
  const int HC  = H * C;        // 256
  const int OUT = 64;           // layer-2 out_dim
  const int N   = in_sizes[0] / IN;       // 50000
  const int E   = in_sizes[1] / 2;        // 800000
  const int Etot = E + N;                 // + self loops

  // ---- workspace layout (floats) ----
  float* ws = (float*)d_ws;
  size_t o = 0;
  float*    h1   = ws + o; o += (size_t)N * HC;   // layer-1 projections
  float*    agg1 = ws + o; o += (size_t)N * HC;   // layer-1 aggregated output
  float*    als1 = ws + o; o += (size_t)N * H;
  float*    ald1 = ws + o; o += (size_t)N * H;
  unsigned* mk1  = (unsigned*)(ws + o); o += (size_t)N * H;
  float*    den1 = ws + o; o += (size_t)N * H;
  float*    h2   = ws + o; o += (size_t)N * OUT;  // layer-2 projections
  float*    als2 = ws + o; o += (size_t)N;
  float*    ald2 = ws + o; o += (size_t)N;
  unsigned* mk2  = (unsigned*)(ws + o); o += (size_t)N;
  float*    den2 = ws + o; o += (size_t)N;
  if (ws_size < o * sizeof(float)) return;

  const unsigned TB = 256;
  const int strips  = N / 16;             // 3125 strips of 16 rows

  // ---- zero all accumulators / stats / output (harness poisons buffers) ----
  gat_zero_f32<<<cdiv((unsigned)(N * HC), TB), TB, 0, stream>>>(agg1, N * HC);
  gat_zero_f32<<<cdiv((unsigned)(N * H),  TB), TB, 0, stream>>>((float*)mk1, N * H);
  gat_zero_f32<<<cdiv((unsigned)(N * H),  TB), TB, 0, stream>>>(den1, N * H);
  gat_zero_f32<<<cdiv((unsigned)N,        TB), TB, 0, stream>>>((float*)mk2, N);
  gat_zero_f32<<<cdiv((unsigned)N,        TB), TB, 0, stream>>>(den2, N);
  gat_zero_f32<<<cdiv((unsigned)(N * OUT),TB), TB, 0, stream>>>(out, N * OUT);

  // ================= Layer 1 =================
  // h1 = x @ W1   (50000x256 @ 256x256)
  {
    dim3 grid(HC / 64, cdiv((unsigned)strips, 4));
    gat_gemm_f32_wmma<256, 256><<<grid, 128, 0, stream>>>(x, W1, h1, N);
  }
  gat_attn_dot<<<cdiv((unsigned)(N * H), TB), TB, 0, stream>>>(h1, as1, ad1, als1, ald1, N, H, C);
  gat_edge_max<<<cdiv((unsigned)(Etot * H), TB), TB, 0, stream>>>(ei, E, Etot, als1, ald1, mk1, H);
  gat_edge_expsum<<<cdiv((unsigned)(Etot * H), TB), TB, 0, stream>>>(ei, E, Etot, als1, ald1, mk1, den1, H);
  gat_edge_aggregate<<<cdiv((unsigned)Etot * 32u, TB), TB, 0, stream>>>(
      ei, E, Etot, als1, ald1, mk1, den1, h1, agg1, H);
  gat_bias_act<<<cdiv((unsigned)(N * HC), TB), TB, 0, stream>>>(agg1, b1, N * HC, HC, 1);

  // ================= Layer 2 =================
  // h2 = agg1 @ W2   (50000x256 @ 256x64)
  {
    dim3 grid(OUT / 64, cdiv((unsigned)strips, 4));
    gat_gemm_f32_wmma<256, 64><<<grid, 128, 0, stream>>>(agg1, W2, h2, N);
  }
  gat_attn_dot<<<cdiv((unsigned)N, TB), TB, 0, stream>>>(h2, as2, ad2, als2, ald2, N, 1, OUT);
  gat_edge_max<<<cdiv((unsigned)Etot, TB), TB, 0, stream>>>(ei, E, Etot, als2, ald2, mk2, 1);
  gat_edge_expsum<<<cdiv((unsigned)Etot, TB), TB, 0, stream>>>(ei, E, Etot, als2, ald2, mk2, den2, 1);
  gat_edge_aggregate<<<cdiv((unsigned)Etot * 32u, TB), TB, 0, stream>>>(
      ei, E, Etot, als2, ald2, mk2, den2, h2, out, 1);
  gat_bias_act<<<cdiv((unsigned)(N * OUT), TB), TB, 0, stream>>>(out, b2, N * OUT, OUT, 0);
}